// DIVLoss_32375463477803
// MI455X (gfx1250) — compile-verified
//
#include <hip/hip_runtime.h>
#include <math.h>

typedef __attribute__((ext_vector_type(2))) float v2f;
typedef __attribute__((ext_vector_type(8))) float v8f;

#define NROWS 4096
#define DIM   1024
#define NCLS  1000

__device__ __forceinline__ float wave_sum(float v) {
#pragma unroll
  for (int off = 16; off > 0; off >>= 1) v += __shfl_xor(v, off, 32);
  return v;
}

// -------------------------------------------------------------------------
// Kernel 1: for each tile of 16 rows, compute raw dots d_raw[i] = f_i . q_{t_i}
// via V_WMMA_F32_16X16X4_F32 (diagonal of A @ gathered-Q^T), and fuse the
// per-row sum-of-squares for 1/||f_i||.
// One wave per 16-row tile; 8 waves / block; 32 blocks.
// A layout (16x4 f32):  lane = 16h+m holds A[m][2h], A[m][2h+1]
// B layout (4x16 f32):  lane = 16h+n holds B[2h][n], B[2h+1][n]  (B = G^T)
// -> both fragments: 2 consecutive floats from row (lane&15) at col kb+2h.
// C layout: VGPR v, lane l -> element (M = v + 8*(l>>4), N = l&15).
// -------------------------------------------------------------------------
__global__ __launch_bounds__(256)
void k_dots_wmma(const float* __restrict__ feature, const float* __restrict__ query,
                 const int* __restrict__ target,
                 float* __restrict__ raw_dot, float* __restrict__ inv_f) {
  const int lane = threadIdx.x & 31;
  const int wave = threadIdx.x >> 5;
  const int m    = lane & 15;
  const int h    = lane >> 4;
  const int tile = blockIdx.x * 8 + wave;          // 0..255
  const int row  = tile * 16 + m;
  const int t    = target[row];

  const float* ar = feature + (size_t)row * DIM + 2 * h;
  const float* br = query   + (size_t)t   * DIM + 2 * h;

  v8f  c  = {};
  float sq = 0.f;
#pragma unroll 4
  for (int kb = 0; kb < DIM; kb += 4) {
    v2f a = *(const v2f*)(ar + kb);
    v2f b = *(const v2f*)(br + kb);
    sq += a.x * a.x + a.y * a.y;
    // 8 args: (neg_a, A, neg_b, B, c_mod, C, reuse_a, reuse_b)
    c = __builtin_amdgcn_wmma_f32_16x16x4_f32(false, a, false, b,
                                              (short)0, c, false, false);
  }

  // full row sum-of-squares: lane m (h=0) covers cols {0,1 mod 4}, lane m+16 the rest
  float sqs = sq + __shfl_xor(sq, 16, 32);
  float ivf = 1.0f / sqrtf(sqs);
  if (h == 0) inv_f[row] = ivf;

  // diagonal element (m,m) lives on this lane iff 0 <= m - 8h < 8
  int vi = m - 8 * h;
  if (vi >= 0 && vi < 8) raw_dot[row] = c[vi];
}

// -------------------------------------------------------------------------
// Kernel 2: partial column sums of normalized feature rows (deterministic).
// 8 blocks x 256 threads; block b accumulates rows [512b, 512b+512).
// -------------------------------------------------------------------------
__global__ __launch_bounds__(256)
void k_fsum_part(const float* __restrict__ feature, const float* __restrict__ inv_f,
                 float* __restrict__ partial) {
  const int col = threadIdx.x * 4;
  const int r0  = blockIdx.x * 512;
  float a0 = 0.f, a1 = 0.f, a2 = 0.f, a3 = 0.f;
  for (int r = r0; r < r0 + 512; ++r) {
    const float iv = inv_f[r];
    const float4 v = *(const float4*)(feature + (size_t)r * DIM + col);
    a0 += v.x * iv; a1 += v.y * iv; a2 += v.z * iv; a3 += v.w * iv;
  }
  *(float4*)(partial + blockIdx.x * DIM + col) = make_float4(a0, a1, a2, a3);
}

__global__ __launch_bounds__(256)
void k_fsum_reduce(const float* __restrict__ partial, float* __restrict__ fsum) {
  const int col = threadIdx.x * 4;
  float a0 = 0.f, a1 = 0.f, a2 = 0.f, a3 = 0.f;
#pragma unroll
  for (int b = 0; b < 8; ++b) {
    const float4 v = *(const float4*)(partial + b * DIM + col);
    a0 += v.x; a1 += v.y; a2 += v.z; a3 += v.w;
  }
  *(float4*)(fsum + col) = make_float4(a0, a1, a2, a3);
}

// -------------------------------------------------------------------------
// Kernel 3: per class c: Sn[c] = (10/4096) * (F_sum . q_c) / ||q_c||, and 1/||q_c||.
// One wave per class; 8 waves/block; 125 blocks (exactly 1000 classes).
// -------------------------------------------------------------------------
__global__ __launch_bounds__(256)
void k_class_sums(const float* __restrict__ query, const float* __restrict__ fsum,
                  float* __restrict__ Sn, float* __restrict__ inv_q) {
  const int lane = threadIdx.x & 31;
  const int wave = threadIdx.x >> 5;
  const int c    = blockIdx.x * 8 + wave;          // 0..999
  const float* qp = query + (size_t)c * DIM;
  float dot = 0.f, sqv = 0.f;
#pragma unroll
  for (int j = 0; j < 8; ++j) {
    const int k = (lane + 32 * j) * 4;
    const float4 q4 = *(const float4*)(qp + k);
    const float4 f4 = *(const float4*)(fsum + k);
    dot += q4.x * f4.x + q4.y * f4.y + q4.z * f4.z + q4.w * f4.w;
    sqv += q4.x * q4.x + q4.y * q4.y + q4.z * q4.z + q4.w * q4.w;
  }
  dot = wave_sum(dot);
  sqv = wave_sum(sqv);
  if (lane == 0) {
    const float iq = 1.0f / sqrtf(sqv);
    inv_q[c] = iq;
    Sn[c]    = dot * iq * (10.0f / 4096.0f);
  }
}

// -------------------------------------------------------------------------
// Kernel 4: stable counting-sort positions. pos[i] = #{j: t_j < t_i} + #{j<i: t_j==t_i}
// (reproduces jnp.argsort(target) stable permutation's inverse).
// Single block, one thread per class, targets staged in LDS.
// -------------------------------------------------------------------------
__global__ __launch_bounds__(1024)
void k_positions(const int* __restrict__ target, int* __restrict__ pos) {
  __shared__ int tg[NROWS];
  for (int i = threadIdx.x; i < NROWS; i += 1024) tg[i] = target[i];
  __syncthreads();
  const int c = threadIdx.x;
  if (c < NCLS) {
    int base = 0;
    for (int i = 0; i < NROWS; ++i) base += (tg[i] < c) ? 1 : 0;
    int k = 0;
    for (int i = 0; i < NROWS; ++i) {
      if (tg[i] == c) { pos[i] = base + k; ++k; }
    }
  }
}

// -------------------------------------------------------------------------
// Kernel 5: loss = (1/4096) * sum_i softplus(100*(Sn[target[pos[i]]] - d[i]))
// with d[i] = raw_dot[i] * inv_f[i] * inv_q[target[i]]. Fixed-order tree reduce.
// -------------------------------------------------------------------------
__global__ __launch_bounds__(256)
void k_loss(const float* __restrict__ raw_dot, const float* __restrict__ inv_f,
            const float* __restrict__ inv_q, const float* __restrict__ Sn,
            const int* __restrict__ pos, const int* __restrict__ target,
            float* __restrict__ out) {
  __shared__ float red[256];
  float acc = 0.f;
#pragma unroll
  for (int j = 0; j < 16; ++j) {
    const int i  = threadIdx.x + 256 * j;
    const int ti = target[i];
    const float d  = raw_dot[i] * inv_f[i] * inv_q[ti];
    const float nn = Sn[target[pos[i]]];
    const float v  = 100.0f * (nn - d);
    acc += fmaxf(v, 0.0f) + log1pf(expf(-fabsf(v)));   // stable softplus
  }
  red[threadIdx.x] = acc;
  __syncthreads();
  for (int s = 128; s > 0; s >>= 1) {
    if (threadIdx.x < s) red[threadIdx.x] += red[threadIdx.x + s];
    __syncthreads();
  }
  if (threadIdx.x == 0) out[0] = red[0] * (1.0f / 4096.0f);
}

// -------------------------------------------------------------------------
extern "C" void kernel_launch(void* const* d_in, const int* in_sizes, int n_in,
                              void* d_out, int out_size, void* d_ws, size_t ws_size,
                              hipStream_t stream) {
  (void)in_sizes; (void)n_in; (void)out_size; (void)ws_size;
  const float* feature = (const float*)d_in[0];   // [4096,1024] f32
  const float* query   = (const float*)d_in[1];   // [1000,1024] f32
  const int*   target  = (const int*)d_in[2];     // [4096] int
  float* out = (float*)d_out;                     // scalar f32

  char* ws = (char*)d_ws;
  float* raw_dot = (float*)(ws + 0);       // 4096 f
  float* inv_f   = (float*)(ws + 16384);   // 4096 f
  float* partial = (float*)(ws + 32768);   // 8*1024 f
  float* fsum    = (float*)(ws + 65536);   // 1024 f
  float* Sn      = (float*)(ws + 69632);   // 1000 f (pad 1024)
  float* inv_q   = (float*)(ws + 73728);   // 1000 f (pad 1024)
  int*   pos     = (int*)  (ws + 77824);   // 4096 i   -> total 94208 B

  k_dots_wmma  <<<32,  256, 0, stream>>>(feature, query, target, raw_dot, inv_f);
  k_fsum_part  <<<8,   256, 0, stream>>>(feature, inv_f, partial);
  k_fsum_reduce<<<1,   256, 0, stream>>>(partial, fsum);
  k_class_sums <<<125, 256, 0, stream>>>(query, fsum, Sn, inv_q);
  k_positions  <<<1,  1024, 0, stream>>>(target, pos);
  k_loss       <<<1,   256, 0, stream>>>(raw_dot, inv_f, inv_q, Sn, pos, target, out);
}